// RackioObserverLSTM_H_34265249088298
// MI455X (gfx1250) — compile-verified
//
#include <hip/hip_runtime.h>
#include <hip/hip_bf16.h>
#include <math.h>

typedef __attribute__((ext_vector_type(2))) float v2f;
typedef __attribute__((ext_vector_type(8))) float v8f;

#define UNITS 1024
#define FEAT  1024
#define GATES 4096   // 4*UNITS

__device__ __forceinline__ float sigf(float x) { return 1.0f / (1.0f + expf(-x)); }

// -------- Kernel A: [xz0; xz1] = X[0:2] @ W + b  via f32 WMMA (M padded to 16) --------
// One wave per 16-wide N tile; K loop in steps of 4 (V_WMMA_F32_16X16X4_F32).
// A loads are UNCONDITIONAL (row pointer clamped to row 0 for pad lanes) and the
// zero-padding is applied with a register select -> no EXEC branches in the loop,
// so the W loads can be claused and pipelined.
__global__ __launch_bounds__(32)
void xw_kernel(const float* __restrict__ X, const float* __restrict__ W,
               const float* __restrict__ bias, float* __restrict__ xz0,
               float* __restrict__ xz1) {
  const int lane = threadIdx.x;      // 0..31
  const int half = lane >> 4;        // selects K sub-pair {0,1} vs {2,3}
  const int l16  = lane & 15;
  const int n0   = blockIdx.x * 16;

  const bool arow = (l16 < 2);                       // only rows 0,1 of A are real
  const float* xp = X + (size_t)(arow ? l16 : 0) * FEAT + 2 * half; // clamped: pad lanes broadcast row 0
  const float* wp = W + (size_t)(2 * half) * GATES + n0 + l16;

  v8f acc = {};
#pragma unroll 8
  for (int k = 0; k < FEAT; k += 4) {
    v2f a, b;
    float vx = xp[k];                // unconditional (b64-fusable) load
    float vy = xp[k + 1];
    a.x = arow ? vx : 0.0f;          // v_cndmask, not a branch
    a.y = arow ? vy : 0.0f;
    b.x = wp[0];                     // W[k + 2*half    ][n0 + l16]
    b.y = wp[GATES];                 // W[k + 2*half + 1][n0 + l16]
    wp += 4 * (size_t)GATES;
    acc = __builtin_amdgcn_wmma_f32_16x16x4_f32(false, a, false, b,
                                                (short)0, acc, false, false);
  }
  if (lane < 16) {                   // D rows: acc[r] lanes0-15 == M=r
    const float bb = bias[n0 + lane];
    xz0[n0 + lane] = acc[0] + bb;    // row M=0 -> timestep 0
    xz1[n0 + lane] = acc[1] + bb;    // row M=1 -> timestep 1
  }
}

// -------- Kernel B: step-0 gating (c0 = 0) --------
__global__ void gate0_kernel(const float* __restrict__ xz0,
                             float* __restrict__ h1, float* __restrict__ c1) {
  int j = blockIdx.x * blockDim.x + threadIdx.x;
  if (j < UNITS) {
    float ig = sigf(xz0[j]);                 // i gate
    float g  = xz0[2 * UNITS + j];           // candidate (identity activation)
    float o  = sigf(xz0[3 * UNITS + j]);     // o gate
    float c  = ig * g;                       // f*c0 == 0
    c1[j] = c;
    h1[j] = o * c;                           // output activation identity
  }
}

// -------- Kernel C: z1 = xz1 + h1 @ U  via f32 WMMA (M padded, only row 0 real) --------
__global__ __launch_bounds__(32)
void hu_kernel(const float* __restrict__ h1, const float* __restrict__ U,
               const float* __restrict__ xz1, float* __restrict__ z1) {
  const int lane = threadIdx.x;
  const int half = lane >> 4;
  const int l16  = lane & 15;
  const int n0   = blockIdx.x * 16;

  const bool arow = (l16 == 0);
  const float* hp = h1 + 2 * half;   // all lanes broadcast the single real row
  const float* up = U + (size_t)(2 * half) * GATES + n0 + l16;

  v8f acc = {};
#pragma unroll 8
  for (int k = 0; k < UNITS; k += 4) {
    v2f a, b;
    float vx = hp[k];                // unconditional broadcast load
    float vy = hp[k + 1];
    a.x = arow ? vx : 0.0f;          // v_cndmask
    a.y = arow ? vy : 0.0f;
    b.x = up[0];
    b.y = up[GATES];
    up += 4 * (size_t)GATES;
    acc = __builtin_amdgcn_wmma_f32_16x16x4_f32(false, a, false, b,
                                                (short)0, acc, false, false);
  }
  if (lane < 16) z1[n0 + lane] = acc[0] + xz1[n0 + lane];
}

// -------- Kernel D: step-1 gating + tanh + dense(Wd,bd,tanh) + final 4 outputs --------
__global__ __launch_bounds__(1024)
void final_kernel(const float* __restrict__ z1, const float* __restrict__ c1,
                  const float* __restrict__ h1, const float* __restrict__ Wd,
                  const float* __restrict__ bd, const float* __restrict__ f,
                  float* __restrict__ out) {
  const int j = threadIdx.x;                 // 0..1023, one unit each
  float ig = sigf(z1[j]);
  float fg = sigf(z1[UNITS + j]);
  float g  = z1[2 * UNITS + j];
  float o  = sigf(z1[3 * UNITS + j]);
  float c2 = fg * c1[j] + ig * g;
  float h2 = o * c2;

  float t0 = tanhf(h1[j]);                   // norm at t=0
  float t1 = tanhf(h2);                      // norm at t=1
  float w0 = Wd[2 * j], w1 = Wd[2 * j + 1];  // Wd is (UNITS, 2) row-major
  float s00 = t0 * w0, s01 = t0 * w1, s10 = t1 * w0, s11 = t1 * w1;

  // wave32 reduction
  for (int off = 16; off > 0; off >>= 1) {
    s00 += __shfl_xor(s00, off, 32);
    s01 += __shfl_xor(s01, off, 32);
    s10 += __shfl_xor(s10, off, 32);
    s11 += __shfl_xor(s11, off, 32);
  }
  __shared__ float r[4][32];
  int wid = j >> 5, lane = j & 31;
  if (lane == 0) { r[0][wid] = s00; r[1][wid] = s01; r[2][wid] = s10; r[3][wid] = s11; }
  __syncthreads();
  if (j == 0) {
    float a0 = 0.f, a1 = 0.f, a2 = 0.f, a3 = 0.f;
    for (int w = 0; w < 32; ++w) { a0 += r[0][w]; a1 += r[1][w]; a2 += r[2][w]; a3 += r[3][w]; }
    float hc00 = tanhf(a0 + bd[0]);
    float hc01 = tanhf(a1 + bd[1]);
    float hc10 = tanhf(a2 + bd[0]);
    float hc11 = tanhf(a3 + bd[1]);
    float den  = f[1] - f[2];                // f is (SEQ,1)
    out[0] = hc00;                           // h_out
    out[1] = hc01;
    out[2] = (hc00 - hc10) / den;            // H
    out[3] = (hc01 - hc11) / den;
  }
}

extern "C" void kernel_launch(void* const* d_in, const int* in_sizes, int n_in,
                              void* d_out, int out_size, void* d_ws, size_t ws_size,
                              hipStream_t stream) {
  const float* X  = (const float*)d_in[0];   // (1, 8192, 1024)
  const float* f  = (const float*)d_in[1];   // (8192, 1)
  const float* W  = (const float*)d_in[2];   // (1024, 4096)
  const float* U  = (const float*)d_in[3];   // (1024, 4096)
  const float* b  = (const float*)d_in[4];   // (4096,)
  const float* Wd = (const float*)d_in[5];   // (1024, 2)
  const float* bd = (const float*)d_in[6];   // (2,)
  float* out = (float*)d_out;                // 4 floats: h_out(2), H(2)

  float* ws  = (float*)d_ws;
  float* xz0 = ws;                 // 4096
  float* xz1 = ws + GATES;         // 4096
  float* h1  = ws + 2 * GATES;     // 1024
  float* c1  = h1 + UNITS;         // 1024
  float* z1  = c1 + UNITS;         // 4096

  xw_kernel   <<<GATES / 16, 32,  0, stream>>>(X, W, b, xz0, xz1);
  gate0_kernel<<<UNITS / 256, 256, 0, stream>>>(xz0, h1, c1);
  hu_kernel   <<<GATES / 16, 32,  0, stream>>>(h1, U, xz1, z1);
  final_kernel<<<1, 1024,     0, stream>>>(z1, c1, h1, Wd, bd, f, out);
}